// GCNLayer_61418032333373
// MI455X (gfx1250) — compile-verified
//
#include <hip/hip_runtime.h>

#define IN_FEATS  128
#define OUT_FEATS 128

typedef float v2f __attribute__((ext_vector_type(2)));
typedef float v8f __attribute__((ext_vector_type(8)));

// ---------------------------------------------------------------------------
// Kernel 1: zero the aggregation workspace (harness poisons d_ws with 0xAA).
// ---------------------------------------------------------------------------
__global__ void gcn_zero_kernel(float4* __restrict__ p, size_t n4) {
    size_t i = (size_t)blockIdx.x * blockDim.x + threadIdx.x;
    size_t stride = (size_t)gridDim.x * blockDim.x;
    for (; i < n4; i += stride) {
        p[i] = make_float4(0.f, 0.f, 0.f, 0.f);
    }
}

// ---------------------------------------------------------------------------
// Kernel 2: edge scatter-add.  One wave per edge-iteration: 32 lanes x float4
// = the full 512B feature row, 4 global_atomic_add_f32 per lane into agg[dst].
// Indices for 32 edges are loaded once per wave and broadcast via __shfl.
// Working set (feature 51MB + agg 51MB + edges 13MB) lives in the 192MB L2.
// ---------------------------------------------------------------------------
__global__ void gcn_scatter_kernel(const float* __restrict__ feat,
                                   const int* __restrict__ src,
                                   const int* __restrict__ dst,
                                   float* __restrict__ agg,
                                   int n_edges) {
    const int lane = threadIdx.x & 31;
    const int wid  = (int)((blockIdx.x * blockDim.x + threadIdx.x) >> 5);
    const int nw   = (int)((gridDim.x * blockDim.x) >> 5);

    for (int base = wid * 32; base < n_edges; base += nw * 32) {
        const int e  = base + lane;
        const int sv = (e < n_edges) ? src[e] : 0;
        const int dv = (e < n_edges) ? dst[e] : 0;
        const int cnt = (n_edges - base) < 32 ? (n_edges - base) : 32;
        for (int i = 0; i < cnt; ++i) {
            const int s = __shfl(sv, i, 32);
            const int d = __shfl(dv, i, 32);
            const float4 v = ((const float4*)(feat + (size_t)s * IN_FEATS))[lane];
            float* ap = agg + (size_t)d * IN_FEATS + lane * 4;
            unsafeAtomicAdd(ap + 0, v.x);
            unsafeAtomicAdd(ap + 1, v.y);
            unsafeAtomicAdd(ap + 2, v.z);
            unsafeAtomicAdd(ap + 3, v.w);
        }
    }
}

// ---------------------------------------------------------------------------
// Kernel 3: out = relu(agg @ W^T + b) via V_WMMA_F32_16X16X4_F32.
//
// Layouts (ISA 7.12.2, 32-bit operands, wave32):
//   A 16x4 : lane L -> row M=L%16, VGPR v holds K = v + 2*(L/16)
//   B 4x16 : lane L -> col N=L%16, VGPR v holds K = v + 2*(L/16)
//   C/D    : lane L -> col N=L%16, VGPR v holds row M = v + 8*(L/16)
// For k-step s, each lane needs A[m, 4s+2g .. 4s+2g+1]  (contiguous float2)
// and B[k,n] = W[n,k] -> W[n, 4s+2g .. 4s+2g+1]          (contiguous float2).
//
// One wave = one 16-row slab: preload 16x128 A into 64 VGPRs (32 x float2),
// then 8 column tiles x 32 WMMAs each with the accumulator held in 8 VGPRs.
// ---------------------------------------------------------------------------
__global__ void __launch_bounds__(128)
gcn_gemm_bias_relu_kernel(const float* __restrict__ agg,
                          const float* __restrict__ W,
                          const float* __restrict__ bias,
                          float* __restrict__ out,
                          int nrows) {
    const int lane  = threadIdx.x & 31;
    const int wave  = threadIdx.x >> 5;               // 0..3
    const int mtile = blockIdx.x * 4 + wave;          // 16-row tile index
    const int mbase = mtile * 16;
    if (mbase >= nrows) return;                       // uniform per wave: EXEC stays all-1s

    const int r = lane & 15;                          // M for A / N for B,D
    const int g = lane >> 4;                          // half-wave select

    // Preload this wave's full A slab row (per-lane: 64 floats = 64 VGPRs).
    v2f a[32];
    const float* arow = agg + (size_t)(mbase + r) * IN_FEATS + 2 * g;
#pragma unroll
    for (int s = 0; s < 32; ++s) {
        a[s] = *(const v2f*)(arow + 4 * s);
    }

#pragma unroll
    for (int t = 0; t < 8; ++t) {                     // 8 column tiles of 16
        v8f acc = {};
        const float* wrow = W + (size_t)(t * 16 + r) * IN_FEATS + 2 * g;
#pragma unroll
        for (int s = 0; s < 32; ++s) {                // K = 128 in steps of 4
            const v2f bv = *(const v2f*)(wrow + 4 * s);
            acc = __builtin_amdgcn_wmma_f32_16x16x4_f32(
                /*neg_a=*/false, a[s], /*neg_b=*/false, bv,
                /*c_mod=*/(short)0, acc, /*reuse_a=*/false, /*reuse_b=*/false);
        }
        const float bb = bias[t * 16 + r];
#pragma unroll
        for (int v = 0; v < 8; ++v) {
            float val = acc[v] + bb;
            val = val > 0.f ? val : 0.f;
            out[(size_t)(mbase + v + 8 * g) * OUT_FEATS + t * 16 + r] = val;
        }
    }
}

// ---------------------------------------------------------------------------
// Launch: zero(agg) -> scatter(atomics) -> wmma gemm+bias+relu, all on stream.
// Inputs: 0=feature f32[N,128], 1=src i32[E], 2=dst i32[E],
//         3=W f32[128,128] (row-major [out,in]), 4=b f32[128].
// ---------------------------------------------------------------------------
extern "C" void kernel_launch(void* const* d_in, const int* in_sizes, int n_in,
                              void* d_out, int out_size, void* d_ws, size_t ws_size,
                              hipStream_t stream) {
    const float* feature = (const float*)d_in[0];
    const int*   src     = (const int*)d_in[1];
    const int*   dst     = (const int*)d_in[2];
    const float* W       = (const float*)d_in[3];
    const float* bias    = (const float*)d_in[4];
    float*       out     = (float*)d_out;

    const int n_nodes = in_sizes[0] / IN_FEATS;
    const int n_edges = in_sizes[1];

    float* agg = (float*)d_ws;                        // [n_nodes, IN_FEATS]
    const size_t agg_elems = (size_t)n_nodes * IN_FEATS;

    // 1) zero agg
    {
        const size_t n4 = agg_elems / 4;
        const int threads = 256;
        int blocks = (int)((n4 + threads - 1) / threads);
        if (blocks > 2048) blocks = 2048;
        gcn_zero_kernel<<<blocks, threads, 0, stream>>>((float4*)agg, n4);
    }

    // 2) scatter-add over edges
    {
        const int threads = 256;                      // 8 waves/block
        const int blocks = 1024;                      // 8192 waves, grid-stride
        gcn_scatter_kernel<<<blocks, threads, 0, stream>>>(feature, src, dst, agg, n_edges);
    }

    // 3) WMMA gemm + bias + relu
    {
        const int mtiles = (n_nodes + 15) / 16;
        const int blocks = (mtiles + 3) / 4;          // 4 waves/block, 1 tile each
        gcn_gemm_bias_relu_kernel<<<blocks, 128, 0, stream>>>(agg, W, bias, out, n_nodes);
    }
}